// loss_22101901705253
// MI455X (gfx1250) — compile-verified
//
#include <hip/hip_runtime.h>
#include <hip/hip_bf16.h>

#define NB      4096      // batch B
#define NW      10        // target words per doc
#define NS      15        // noise per target
#define ND      128       // embedding dim
#define NT      32        // topics
#define ROWS_PB 16        // b-rows per k1 block
#define K1_BLOCKS (NB / ROWS_PB)            // 256
#define PAIRS   (NB * NW)                   // 40960
#define K2_BLOCKS (PAIRS / 8)               // 5120, 8 waves/block
#define LOGEPS  (-20.72326583694641f)       // ln(1e-9)
#define DIR_SCALE 96.875f                   // LAMBDA*(1-ALPHA) = 100*(31/32)

typedef __attribute__((ext_vector_type(2))) float v2f;
typedef __attribute__((ext_vector_type(8))) float v8f;

__device__ __forceinline__ float logsig_clipped(float x) {
    // log(max(sigmoid(x), 1e-9)) = max(-softplus(-x), ln(1e-9)), numerically stable
    float t  = -x;
    float sp = fmaxf(t, 0.0f) + __logf(1.0f + __expf(-fabsf(t)));
    return fmaxf(-sp, LOGEPS);
}

// ---------------------------------------------------------------------------
// Kernel 1: per-doc softmax, doc_vectors = probs @ topic_vectors via
// V_WMMA_F32_16X16X4_F32 (full f32 precision), context = doc_vectors + pivot.
// Also emits per-block partials for Sw and the dirichlet sum S2.
// ---------------------------------------------------------------------------
__global__ void k1_context(const int* __restrict__ doc_indices,
                           const int* __restrict__ pivot_words,
                           const float* __restrict__ word_vectors,
                           const float* __restrict__ topic_vectors,
                           const float* __restrict__ doc_weights_table,
                           const float* __restrict__ loss_w_all,
                           float* __restrict__ ctx,
                           float* __restrict__ w_un,
                           float* __restrict__ part2,
                           float* __restrict__ partW) {
    __shared__ float p[ROWS_PB][NT + 1];   // softmax probs, padded stride
    __shared__ int   piv[ROWS_PB];
    __shared__ float rowW[ROWS_PB], rowS2[ROWS_PB];

    const int tid = threadIdx.x;
    const int b0  = blockIdx.x * ROWS_PB;

    if (tid < ROWS_PB) {
        const int b  = b0 + tid;
        const int di = doc_indices[b];
        piv[tid] = pivot_words[b];
        const float wu = loss_w_all[di];
        w_un[b]   = wu;
        rowW[tid] = wu;

        const float* dwp = doc_weights_table + (size_t)di * NT;
        float dw[NT];
        float m = -1e30f, sum_dw = 0.0f;
#pragma unroll
        for (int k = 0; k < NT; ++k) {
            dw[k] = dwp[k];
            sum_dw += dw[k];
            m = fmaxf(m, dw[k]);
        }
        float s = 0.0f;
#pragma unroll
        for (int k = 0; k < NT; ++k) {
            float e = __expf(dw[k] - m);
            s += e;
            p[tid][k] = e;
        }
        const float inv = 1.0f / s;
#pragma unroll
        for (int k = 0; k < NT; ++k) p[tid][k] *= inv;

        // sum_k log_softmax = sum_dw - T*max - T*log(sumexp)
        rowS2[tid] = wu * (sum_dw - (float)NT * m - (float)NT * __logf(s));
    }
    __syncthreads();

    if (tid == 0) {
        float a = 0.0f, ww = 0.0f;
        for (int i = 0; i < ROWS_PB; ++i) { a += rowS2[i]; ww += rowW[i]; }
        part2[blockIdx.x] = a;
        partW[blockIdx.x] = ww;
    }

    // ---- WMMA phase: 8 waves, wave w covers output cols [16w, 16w+16) ----
    const int lane = tid & 31;
    const int wave = tid >> 5;
    const int half = lane >> 4;            // 0: lanes 0-15, 1: lanes 16-31
    const int colN = wave * 16 + (lane & 15);
    const int rowA = lane & 15;
    const int koff = half * 2;

    v8f acc = {};
#pragma unroll
    for (int k = 0; k < NT / 4; ++k) {     // 8 steps of K=4
        const int kk = k * 4 + koff;
        v2f a, bb;
        a[0]  = p[rowA][kk];
        a[1]  = p[rowA][kk + 1];
        bb[0] = topic_vectors[(size_t)kk * ND + colN];
        bb[1] = topic_vectors[(size_t)(kk + 1) * ND + colN];
        acc = __builtin_amdgcn_wmma_f32_16x16x4_f32(
            /*neg_a=*/false, a, /*neg_b=*/false, bb,
            /*c_mod=*/(short)0, acc, /*reuse_a=*/false, /*reuse_b=*/false);
    }

    // D layout: VGPR r holds M=r (lanes 0-15) / M=r+8 (lanes 16-31), N=colN
#pragma unroll
    for (int r = 0; r < 8; ++r) {
        const int M = half * 8 + r;
        const int b = b0 + M;
        const float v = acc[r] + word_vectors[(size_t)piv[M] * ND + colN];
        ctx[(size_t)b * ND + colN] = v;
    }
}

// ---------------------------------------------------------------------------
// Kernel 2: one wave per (b,w) pair. Wave-uniform gather indices (scalar
// loads), float4 gathers, 128-dim dots vs context, register-halving wave32
// reduction (21 shuffles/wave), clipped log-sigmoid, deterministic partials.
// ---------------------------------------------------------------------------
__global__ void k2_loss(const int* __restrict__ target_words,
                        const int* __restrict__ noise_words,
                        const float* __restrict__ word_vectors,
                        const float* __restrict__ ctx,
                        const float* __restrict__ w_un,
                        float* __restrict__ part1) {
    __shared__ float wacc[8];
    const int tid  = threadIdx.x;
    const int lane = tid & 31;
    const int wave = tid >> 5;
    // pair is wave-uniform: force it into an SGPR so index gathers go scalar
    const int pair = __builtin_amdgcn_readfirstlane(blockIdx.x * 8 + wave);
    const int b    = pair / NW;
    const int w    = pair - b * NW;

    int rows[16];
    rows[0] = target_words[b * NW + w];
#pragma unroll
    for (int s = 0; s < NS; ++s) rows[s + 1] = noise_words[pair * NS + s];

    const float4 c = ((const float4*)(ctx + (size_t)b * ND))[lane];

    float a[16];
#pragma unroll
    for (int j = 0; j < 16; ++j) {
        const float4 v = ((const float4*)(word_vectors + (size_t)rows[j] * ND))[lane];
        a[j] = v.x * c.x + v.y * c.y + v.z * c.z + v.w * c.w;
    }

    // Register-halving reduction: word bit3<-lane4, bit2<-lane3, bit1<-lane2,
    // bit0<-lane1.  8+4+2+1 shuffles to fold 16 regs -> 1, +1 for lane bit0.
    float r8[8], r4[4], r2[2], r1;
    {
        const bool hi = (lane & 16) != 0;
#pragma unroll
        for (int j = 0; j < 8; ++j) {
            const float send = hi ? a[j] : a[j + 8];
            const float keep = hi ? a[j + 8] : a[j];
            r8[j] = keep + __shfl_xor(send, 16, 32);
        }
    }
    {
        const bool hi = (lane & 8) != 0;
#pragma unroll
        for (int j = 0; j < 4; ++j) {
            const float send = hi ? r8[j] : r8[j + 4];
            const float keep = hi ? r8[j + 4] : r8[j];
            r4[j] = keep + __shfl_xor(send, 8, 32);
        }
    }
    {
        const bool hi = (lane & 4) != 0;
#pragma unroll
        for (int j = 0; j < 2; ++j) {
            const float send = hi ? r4[j] : r4[j + 2];
            const float keep = hi ? r4[j + 2] : r4[j];
            r2[j] = keep + __shfl_xor(send, 4, 32);
        }
    }
    {
        const bool hi = (lane & 2) != 0;
        const float send = hi ? r2[0] : r2[1];
        const float keep = hi ? r2[1] : r2[0];
        r1 = keep + __shfl_xor(send, 2, 32);
    }
    r1 += __shfl_xor(r1, 1, 32);           // fold remaining lane bit0

    // even lane L now holds the full dot for word (L>>1)&15 (word 0 = target)
    const int wordid = (lane >> 1) & 15;
    float contrib = 0.0f;
    if ((lane & 1) == 0)
        contrib = logsig_clipped(wordid == 0 ? r1 : -r1);
#pragma unroll
    for (int off = 16; off >= 1; off >>= 1)
        contrib += __shfl_xor(contrib, off, 32);

    if (lane == 0) wacc[wave] = w_un[b] * contrib;
    __syncthreads();
    if (tid == 0) {
        float s = 0.0f;
        for (int i = 0; i < 8; ++i) s += wacc[i];
        part1[blockIdx.x] = s;
    }
}

// ---------------------------------------------------------------------------
// Kernel 3: deterministic final reduction + output scalars.
// ---------------------------------------------------------------------------
__global__ void k3_final(const float* __restrict__ part1,
                         const float* __restrict__ part2,
                         const float* __restrict__ partW,
                         float* __restrict__ out) {
    __shared__ float sm[256];
    const int tid = threadIdx.x;

    float a = 0.0f;
    for (int i = tid; i < K2_BLOCKS; i += 256) a += part1[i];
    sm[tid] = a; __syncthreads();
    for (int s = 128; s > 0; s >>= 1) { if (tid < s) sm[tid] += sm[tid + s]; __syncthreads(); }
    const float S1 = sm[0]; __syncthreads();

    sm[tid] = (tid < K1_BLOCKS) ? part2[tid] : 0.0f; __syncthreads();
    for (int s = 128; s > 0; s >>= 1) { if (tid < s) sm[tid] += sm[tid + s]; __syncthreads(); }
    const float S2 = sm[0]; __syncthreads();

    sm[tid] = (tid < K1_BLOCKS) ? partW[tid] : 0.0f; __syncthreads();
    for (int s = 128; s > 0; s >>= 1) { if (tid < s) sm[tid] += sm[tid + s]; __syncthreads(); }
    const float Sw = sm[0];

    if (tid == 0) {
        out[0] = -S1 / Sw;               // neg_loss = -(1/Sw) * sum(w_u * X)
        out[1] = DIR_SCALE * S2 / Sw;    // dirichlet
    }
}

extern "C" void kernel_launch(void* const* d_in, const int* in_sizes, int n_in,
                              void* d_out, int out_size, void* d_ws, size_t ws_size,
                              hipStream_t stream) {
    const int*   doc_indices       = (const int*)d_in[0];
    const int*   pivot_words       = (const int*)d_in[1];
    const int*   target_words      = (const int*)d_in[2];
    const int*   noise_words       = (const int*)d_in[3];
    const float* word_vectors      = (const float*)d_in[4];
    const float* topic_vectors     = (const float*)d_in[5];
    const float* doc_weights_table = (const float*)d_in[6];
    const float* loss_w_all        = (const float*)d_in[7];
    float* out = (float*)d_out;

    float* ws    = (float*)d_ws;
    float* ctx   = ws;                         // NB*ND
    float* w_un  = ctx + (size_t)NB * ND;      // NB
    float* part2 = w_un + NB;                  // K1_BLOCKS
    float* partW = part2 + K1_BLOCKS;          // K1_BLOCKS
    float* part1 = partW + K1_BLOCKS;          // K2_BLOCKS

    k1_context<<<K1_BLOCKS, 256, 0, stream>>>(doc_indices, pivot_words, word_vectors,
                                              topic_vectors, doc_weights_table, loss_w_all,
                                              ctx, w_un, part2, partW);
    k2_loss<<<K2_BLOCKS, 256, 0, stream>>>(target_words, noise_words, word_vectors,
                                           ctx, w_un, part1);
    k3_final<<<1, 256, 0, stream>>>(part1, part2, partW, out);
}